// GATNet_19713899889121
// MI455X (gfx1250) — compile-verified
//
#include <hip/hip_runtime.h>
#include <hip/hip_bf16.h>

#define N_NODES 50000
#define N_EDGES 800000
#define F_IN 128
#define HID 64
#define EDIM 16
#define NGRAPH 64
#define NEG_SLOPE 0.2f

typedef float v2f __attribute__((ext_vector_type(2)));
typedef float v8f __attribute__((ext_vector_type(8)));

// ---------------------------------------------------------------------------
// Y = X[M,K] @ W[K,64] for two weight matrices at once (xl = X@Wl, xr = X@Wr).
// One wave computes one 16x16 output tile via V_WMMA_F32_16X16X4_F32.
// Block = 256 threads = 8 waves: waves 0..3 -> xl n-tiles, waves 4..7 -> xr.
// A-frag layout (16x4 f32): lanes 0-15 rows M=0..15; VGPR0={K=k0,K=k0+2},
// VGPR1={K=k0+1,K=k0+3} selected by lane[4]. B-frag (4x16) mirrored on K.
// C/D layout: VGPR v -> M = v + 8*lane[4], N = lane[3:0].
// ---------------------------------------------------------------------------
__global__ void gemm_two_wmma(const float* __restrict__ X, int M, int K,
                              const float* __restrict__ Wl,
                              const float* __restrict__ Wr,
                              float* __restrict__ xl, float* __restrict__ xr) {
    const int wave = threadIdx.x >> 5;
    const int lane = threadIdx.x & 31;
    const int mbase = blockIdx.x * 16;
    if (mbase >= M) return;
    const float* __restrict__ W = (wave < 4) ? Wl : Wr;
    float* __restrict__ Y = (wave < 4) ? xl : xr;
    const int nbase = (wave & 3) * 16;
    const int r = lane & 15;           // row within A-tile / col within B-tile
    const int khalf = (lane >> 4) * 2; // 0 or 2

    v8f acc = {};
    for (int k = 0; k < K; k += 4) {
        v2f a, b;
        const int m = mbase + r;
        a.x = X[m * K + k + khalf];
        a.y = X[m * K + k + khalf + 1];
        b.x = W[(k + khalf) * HID + nbase + r];
        b.y = W[(k + khalf + 1) * HID + nbase + r];
        acc = __builtin_amdgcn_wmma_f32_16x16x4_f32(
            /*neg_a=*/false, a, /*neg_b=*/false, b,
            /*c_mod=*/(short)0, acc, /*reuse_a=*/false, /*reuse_b=*/false);
    }
    const int mro = (lane >> 4) * 8;
#pragma unroll
    for (int v = 0; v < 8; ++v) {
        Y[(mbase + v + mro) * HID + nbase + r] = acc[v];
    }
}

// ---------------------------------------------------------------------------
// CSR-by-destination build (once per call): degree count, exclusive scan,
// edge-id scatter. Integer atomics only, one-time cost.
// ---------------------------------------------------------------------------
__global__ void fill_f32(float* __restrict__ p, float v, int n) {
    int i = blockIdx.x * blockDim.x + threadIdx.x;
    if (i < n) p[i] = v;
}

__global__ void fill_i32(int* __restrict__ p, int v, int n) {
    int i = blockIdx.x * blockDim.x + threadIdx.x;
    if (i < n) p[i] = v;
}

__global__ void count_deg(const int* __restrict__ dst, int* __restrict__ deg, int E) {
    int j = blockIdx.x * blockDim.x + threadIdx.x;
    if (j < E) atomicAdd(&deg[dst[j]], 1);
}

// single 1024-thread block: hierarchical exclusive scan of deg[0..n) into
// rowptr[0..n] (rowptr[n] = E), and a second copy into cursor for scatter.
__global__ void scan_deg(const int* __restrict__ deg, int* __restrict__ rowptr,
                         int* __restrict__ cursor, int n) {
    __shared__ int part[1024];
    const int T = 1024;
    const int tid = threadIdx.x;
    const int chunk = (n + T - 1) / T;
    const int beg = tid * chunk;
    const int end = min(beg + chunk, n);
    int s = 0;
    for (int i = beg; i < end; ++i) s += deg[i];
    part[tid] = s;
    __syncthreads();
    // Hillis-Steele inclusive scan over the 1024 partials
    for (int off = 1; off < T; off <<= 1) {
        int v = (tid >= off) ? part[tid - off] : 0;
        __syncthreads();
        part[tid] += v;
        __syncthreads();
    }
    int run = (tid == 0) ? 0 : part[tid - 1];
    for (int i = beg; i < end; ++i) {
        rowptr[i] = run;
        cursor[i] = run;
        run += deg[i];
    }
    if (tid == 0) rowptr[n] = part[T - 1];
}

__global__ void scatter_eid(const int* __restrict__ dst, int* __restrict__ cursor,
                            int* __restrict__ eid, int E) {
    int j = blockIdx.x * blockDim.x + threadIdx.x;
    if (j >= E) return;
    int pos = atomicAdd(&cursor[dst[j]], 1);
    eid[pos] = j;
}

// ---------------------------------------------------------------------------
// Edge score: one wave per edge. Fused edge transform e = edge_attr[j] @ We
// (We staged in LDS), score = att . leaky_relu(xl[src]+xr[dst]+e).
// No atomics: score is consumed by the gather-formulated softmax below.
// ---------------------------------------------------------------------------
__global__ void edge_score(const float* __restrict__ xl,
                           const float* __restrict__ xr,
                           const float* __restrict__ eattr,
                           const float* __restrict__ We,
                           const float* __restrict__ att,
                           const int* __restrict__ src,
                           const int* __restrict__ dst,
                           float* __restrict__ score, int E) {
    __shared__ float sWe[EDIM * HID];
    __shared__ float sAtt[HID];
    for (int i = threadIdx.x; i < EDIM * HID; i += blockDim.x) sWe[i] = We[i];
    for (int i = threadIdx.x; i < HID; i += blockDim.x) sAtt[i] = att[i];
    __syncthreads();

    const int wave = threadIdx.x >> 5;
    const int lane = threadIdx.x & 31;
    const int j = blockIdx.x * 8 + wave;
    if (j >= E) return;
    const int s = src[j], d = dst[j];

    float ea[EDIM];
#pragma unroll
    for (int k = 0; k < EDIM; ++k) ea[k] = eattr[j * EDIM + k];

    float t = 0.0f;
#pragma unroll
    for (int half = 0; half < 2; ++half) {
        const int c = lane + half * 32;
        float e = 0.0f;
#pragma unroll
        for (int k = 0; k < EDIM; ++k) e += ea[k] * sWe[k * HID + c];
        const float pre = xl[s * HID + c] + xr[d * HID + c] + e;
        const float lr = pre > 0.0f ? pre : NEG_SLOPE * pre;
        t += lr * sAtt[c];
    }
#pragma unroll
    for (int off = 16; off > 0; off >>= 1) t += __shfl_xor(t, off, 32);
    if (lane == 0) score[j] = t;
}

// ---------------------------------------------------------------------------
// Gather-formulated segment softmax + aggregation + bias + ELU.
// One wave per destination node: wave32 shfl_xor reductions for max / sum,
// then register accumulation of sum(alpha * xl[src]) with lane = channel
// (coalesced 64-float row gathers out of L2). Zero atomics.
// ---------------------------------------------------------------------------
__global__ void node_softmax_agg(const float* __restrict__ xl,
                                 const float* __restrict__ score,
                                 const int* __restrict__ rowptr,
                                 const int* __restrict__ eid,
                                 const int* __restrict__ src,
                                 const float* __restrict__ b,
                                 float* __restrict__ h, int n) {
    const int wave = threadIdx.x >> 5;
    const int lane = threadIdx.x & 31;
    const int i = blockIdx.x * 8 + wave;
    if (i >= n) return;
    const int beg = rowptr[i], end = rowptr[i + 1];

    float mx = -INFINITY;
    for (int t = beg + lane; t < end; t += 32) mx = fmaxf(mx, score[eid[t]]);
#pragma unroll
    for (int off = 16; off > 0; off >>= 1) mx = fmaxf(mx, __shfl_xor(mx, off, 32));

    float sum = 0.0f;
    for (int t = beg + lane; t < end; t += 32) sum += __expf(score[eid[t]] - mx);
#pragma unroll
    for (int off = 16; off > 0; off >>= 1) sum += __shfl_xor(sum, off, 32);
    const float inv = 1.0f / (sum + 1e-16f);

    float acc0 = 0.0f, acc1 = 0.0f;
    for (int t = beg; t < end; ++t) {
        const int j = eid[t];                          // uniform -> broadcast load
        const float alpha = __expf(score[j] - mx) * inv;
        const int s = src[j];
        acc0 += alpha * xl[s * HID + lane];
        acc1 += alpha * xl[s * HID + lane + 32];
    }
    const float v0 = acc0 + b[lane];
    const float v1 = acc1 + b[lane + 32];
    h[i * HID + lane]      = v0 > 0.0f ? v0 : (__expf(v0) - 1.0f);
    h[i * HID + lane + 32] = v1 > 0.0f ? v1 : (__expf(v1) - 1.0f);
}

// mean-pool accumulation: sums[batch[i]] += h[i], cnt[batch[i]] += 1 (one-time)
__global__ void pool_sum(const float* __restrict__ h,
                         const int* __restrict__ batch,
                         float* __restrict__ sums,
                         float* __restrict__ cnt, int n) {
    int idx = blockIdx.x * blockDim.x + threadIdx.x;
    if (idx >= n) return;
    const int i = idx >> 6, c = idx & 63;
    const int g = batch[i];
    atomicAdd(&sums[g * HID + c], h[idx]);
    if (c == 0) atomicAdd(&cnt[g], 1.0f);
}

// MLP head [64,64,32,16,1]; single block of 64 threads, one graph per thread
__global__ void mlp_head(const float* __restrict__ sums,
                         const float* __restrict__ cnt,
                         const float* __restrict__ M1, const float* __restrict__ mb1,
                         const float* __restrict__ M2, const float* __restrict__ mb2,
                         const float* __restrict__ M3, const float* __restrict__ mb3,
                         const float* __restrict__ M4, const float* __restrict__ mb4,
                         float* __restrict__ out) {
    const int g = threadIdx.x;
    if (g >= NGRAPH) return;
    float a[64], t[64];
    float c = cnt[g];
    if (c < 1.0f) c = 1.0f;
    for (int k = 0; k < 64; ++k) a[k] = sums[g * HID + k] / c;
    for (int o = 0; o < 64; ++o) {
        float acc = mb1[o];
        for (int k = 0; k < 64; ++k) acc += a[k] * M1[k * 64 + o];
        t[o] = fmaxf(acc, 0.0f);
    }
    for (int o = 0; o < 32; ++o) {
        float acc = mb2[o];
        for (int k = 0; k < 64; ++k) acc += t[k] * M2[k * 32 + o];
        a[o] = fmaxf(acc, 0.0f);
    }
    for (int o = 0; o < 16; ++o) {
        float acc = mb3[o];
        for (int k = 0; k < 32; ++k) acc += a[k] * M3[k * 16 + o];
        t[o] = fmaxf(acc, 0.0f);
    }
    float acc = mb4[0];
    for (int k = 0; k < 16; ++k) acc += t[k] * M4[k];
    out[g] = acc;
}

// ---------------------------------------------------------------------------
extern "C" void kernel_launch(void* const* d_in, const int* in_sizes, int n_in,
                              void* d_out, int out_size, void* d_ws, size_t ws_size,
                              hipStream_t stream) {
    const float* x         = (const float*)d_in[0];
    const float* edge_attr = (const float*)d_in[1];
    const int*   edge_idx  = (const int*)d_in[2];
    const int*   batch     = (const int*)d_in[3];
    const float* Wl0 = (const float*)d_in[4];
    const float* Wr0 = (const float*)d_in[5];
    const float* We0 = (const float*)d_in[6];
    const float* att0 = (const float*)d_in[7];
    const float* b0   = (const float*)d_in[8];
    const float* Wl   = (const float*)d_in[9];
    const float* Wr   = (const float*)d_in[10];
    const float* We   = (const float*)d_in[11];
    const float* att  = (const float*)d_in[12];
    const float* b    = (const float*)d_in[13];
    const float* M1 = (const float*)d_in[14]; const float* mb1 = (const float*)d_in[15];
    const float* M2 = (const float*)d_in[16]; const float* mb2 = (const float*)d_in[17];
    const float* M3 = (const float*)d_in[18]; const float* mb3 = (const float*)d_in[19];
    const float* M4 = (const float*)d_in[20]; const float* mb4 = (const float*)d_in[21];

    const int* src = edge_idx;            // edge_index[0]
    const int* dst = edge_idx + N_EDGES;  // edge_index[1]

    // workspace carve-up
    char* ws = (char*)d_ws;
    float* h      = (float*)ws; ws += (size_t)N_NODES * HID * 4;
    float* xl     = (float*)ws; ws += (size_t)N_NODES * HID * 4;
    float* xr     = (float*)ws; ws += (size_t)N_NODES * HID * 4;
    float* score  = (float*)ws; ws += (size_t)N_EDGES * 4;
    float* sums   = (float*)ws; ws += (size_t)NGRAPH * HID * 4;
    float* cnt    = (float*)ws; ws += (size_t)NGRAPH * 4;
    int*   deg    = (int*)ws;   ws += (size_t)N_NODES * 4;
    int*   cursor = (int*)ws;   ws += (size_t)N_NODES * 4;
    int*   rowptr = (int*)ws;   ws += (size_t)(N_NODES + 1) * 4;
    int*   eid    = (int*)ws;   ws += (size_t)N_EDGES * 4;

    const int NB = 256;
    const int gNodeC = (N_NODES * HID + NB - 1) / NB;
    const int gNode  = (N_NODES + NB - 1) / NB;
    const int gEdge  = (N_EDGES + NB - 1) / NB;
    const int gEdgeW = (N_EDGES + 7) / 8;   // one wave per edge
    const int gNodeW = (N_NODES + 7) / 8;   // one wave per node
    const int gMtile = (N_NODES + 15) / 16; // 16-row WMMA tiles

    // one-time CSR-by-destination build (re-done every call: no cached state)
    fill_i32<<<gNode, NB, 0, stream>>>(deg, 0, N_NODES);
    count_deg<<<gEdge, NB, 0, stream>>>(dst, deg, N_EDGES);
    scan_deg<<<1, 1024, 0, stream>>>(deg, rowptr, cursor, N_NODES);
    scatter_eid<<<gEdge, NB, 0, stream>>>(dst, cursor, eid, N_EDGES);

    for (int layer = 0; layer < 4; ++layer) {
        const float* hin  = (layer == 0) ? x : h;
        const int    K    = (layer == 0) ? F_IN : HID;
        const float* Wl_l = (layer == 0) ? Wl0 : Wl + (size_t)(layer - 1) * HID * HID;
        const float* Wr_l = (layer == 0) ? Wr0 : Wr + (size_t)(layer - 1) * HID * HID;
        const float* We_l = (layer == 0) ? We0 : We + (size_t)(layer - 1) * EDIM * HID;
        const float* at_l = (layer == 0) ? att0 : att + (size_t)(layer - 1) * HID;
        const float* b_l  = (layer == 0) ? b0 : b + (size_t)(layer - 1) * HID;

        gemm_two_wmma<<<gMtile, 256, 0, stream>>>(hin, N_NODES, K, Wl_l, Wr_l, xl, xr);
        edge_score<<<gEdgeW, 256, 0, stream>>>(xl, xr, edge_attr, We_l, at_l,
                                               src, dst, score, N_EDGES);
        node_softmax_agg<<<gNodeW, 256, 0, stream>>>(xl, score, rowptr, eid, src,
                                                     b_l, h, N_NODES);
    }

    fill_f32<<<(NGRAPH * HID + NB - 1) / NB, NB, 0, stream>>>(sums, 0.0f, NGRAPH * HID);
    fill_f32<<<1, NB, 0, stream>>>(cnt, 0.0f, NGRAPH);
    pool_sum<<<gNodeC, NB, 0, stream>>>(h, batch, sums, cnt, N_NODES * HID);
    mlp_head<<<1, 64, 0, stream>>>(sums, cnt, M1, mb1, M2, mb2, M3, mb3, M4, mb4,
                                   (float*)d_out);
}